// Detect_4389456576981
// MI455X (gfx1250) — compile-verified
//
#include <hip/hip_runtime.h>
#include <hip/hip_bf16.h>
#include <math.h>

#define BATCH   32
#define PRIORS  16384
#define NCLS    81
#define TOPK    200
#define THREADS 256
#define NWAVES  (THREADS / 32)
#define CHUNK   (PRIORS / THREADS)   // 64 scores per thread
#define CONF_T  0.01f
#define NMS_T   0.45f

// ---- CDNA5 async global->LDS path (guarded; falls back to plain loads) ----
#if defined(__HIP_DEVICE_COMPILE__) && defined(__gfx1250__)
#if __has_builtin(__builtin_amdgcn_global_load_async_to_lds_b32) && \
    __has_builtin(__builtin_amdgcn_s_wait_asynccnt)
#define USE_ASYNC_LDS 1
#endif
#endif

typedef __attribute__((address_space(1))) int* gptr_i32;
typedef __attribute__((address_space(3))) int* lptr_i32;

// ---------------------------------------------------------------------------
// Kernel 0: zero the output (harness poisons d_out; class 0 + tail rows must
// be exactly 0.0 and we only write kept detections).
// ---------------------------------------------------------------------------
__global__ void zero_out_kernel(float* __restrict__ o, int n) {
  int i = blockIdx.x * blockDim.x + threadIdx.x;
  int stride = gridDim.x * blockDim.x;
  for (; i < n; i += stride) o[i] = 0.0f;
}

// ---------------------------------------------------------------------------
// Kernel 1: SSD box decode.  boxes[b*P+p] = (xmin, ymin, xmax, ymax)
// ---------------------------------------------------------------------------
__global__ __launch_bounds__(256)
void decode_kernel(const float* __restrict__ loc,
                   const float* __restrict__ prior,
                   float* __restrict__ boxes) {
  int idx = blockIdx.x * 256 + threadIdx.x;
  if (idx >= BATCH * PRIORS) return;
  int p = idx & (PRIORS - 1);
  float4 l  = reinterpret_cast<const float4*>(loc)[idx];
  float4 pr = reinterpret_cast<const float4*>(prior)[p];
  float cx = pr.x + l.x * 0.1f * pr.z;
  float cy = pr.y + l.y * 0.1f * pr.w;
  float w  = pr.z * expf(l.z * 0.2f);
  float h  = pr.w * expf(l.w * 0.2f);
  float4 o;
  o.x = cx - 0.5f * w;
  o.y = cy - 0.5f * h;
  o.z = o.x + w;
  o.w = o.y + h;
  reinterpret_cast<float4*>(boxes)[idx] = o;
}

// ---------------------------------------------------------------------------
// Kernel 2: per-(image, class) top-K + greedy NMS + stable compaction.
// One 256-thread block (8 wave32s) per (b, c) pair; c in [1, 80].
// ---------------------------------------------------------------------------
__global__ __launch_bounds__(THREADS)
void detect_kernel(const float* __restrict__ conf,   // [B*P, C]
                   const float* __restrict__ boxes,  // [B*P, 4] (ws)
                   float* __restrict__ out) {        // [B, C, K, 5]
  const int tid  = threadIdx.x;
  const int lane = tid & 31;
  const int wave = tid >> 5;
  const int b    = blockIdx.x / (NCLS - 1);
  const int c    = blockIdx.x % (NCLS - 1) + 1;

  extern __shared__ float smem[];
  float*              s       = smem;                                   // [16384]
  unsigned long long* wavemax = (unsigned long long*)(smem + PRIORS);   // [8]
  unsigned long long* winner  = wavemax + NWAVES;                       // [1]
  float* cand_s = (float*)(winner + 1);                                 // [200]
  int*   cand_i = (int*)  (cand_s + TOPK);                              // [200]
  float* bx     = (float*)(cand_i + TOPK);                              // [200*4]
  float* area   = bx + TOPK * 4;                                        // [200]
  int*   keep   = (int*)(area + TOPK);                                  // [200]
  int*   pos    = keep + TOPK;                                          // [200]

  // ---- Stage this class's score column into LDS: s[p], p = i*256 + tid ----
  const float* colbase = conf + (size_t)b * PRIORS * NCLS + c;
#if defined(USE_ASYNC_LDS)
  #pragma unroll 4
  for (int i = 0; i < CHUNK; ++i) {
    int p = i * THREADS + tid;
    __builtin_amdgcn_global_load_async_to_lds_b32(
        (gptr_i32)(colbase + (size_t)p * NCLS),
        (lptr_i32)(s + p),
        0, 0);
  }
  __builtin_amdgcn_s_wait_asynccnt(0);
  __syncthreads();
#else
  #pragma unroll 4
  for (int i = 0; i < CHUNK; ++i) {
    int p = i * THREADS + tid;
    s[p] = colbase[(size_t)p * NCLS];
  }
  __syncthreads();
#endif

  // ---- Threshold in place; build per-thread running max key. Key packs
  // (score_bits << 32) | (0xFFFFFFFF - p): max key == highest score, ties
  // broken toward the lowest prior index (matches jax.lax.top_k). ----
  unsigned long long curkey = 0ULL;
  for (int i = 0; i < CHUNK; ++i) {
    int p = i * THREADS + tid;
    float v = s[p];
    if (v > CONF_T) {
      unsigned long long key =
          ((unsigned long long)__float_as_uint(v) << 32) |
          (unsigned long long)(0xFFFFFFFFu - (unsigned)p);
      if (key > curkey) curkey = key;
    } else {
      s[p] = -__builtin_inff();
    }
  }
  __syncthreads();

  // ---- 200 extraction rounds ----
  for (int k = 0; k < TOPK; ++k) {
    unsigned long long w = curkey;
    #pragma unroll
    for (int off = 16; off > 0; off >>= 1) {
      unsigned long long o = __shfl_xor(w, off, 32);
      if (o > w) w = o;
    }
    if (lane == 0) wavemax[wave] = w;
    __syncthreads();
    if (tid == 0) {
      unsigned long long m = wavemax[0];
      #pragma unroll
      for (int q = 1; q < NWAVES; ++q)
        if (wavemax[q] > m) m = wavemax[q];
      *winner = m;
    }
    __syncthreads();
    unsigned long long win = *winner;
    if (tid == 0) {
      if (win != 0ULL) {
        cand_s[k] = __uint_as_float((unsigned)(win >> 32));
        cand_i[k] = (int)(0xFFFFFFFFu - (unsigned)(win & 0xFFFFFFFFull));
      } else {
        cand_s[k] = -__builtin_inff();
        cand_i[k] = -1;
      }
    }
    if (win != 0ULL && curkey == win) {
      // unique owner: remove winner from my chunk, rescan my 64 LDS slots
      int p = (int)(0xFFFFFFFFu - (unsigned)(win & 0xFFFFFFFFull));
      s[p] = -__builtin_inff();
      unsigned long long nk = 0ULL;
      for (int i = 0; i < CHUNK; ++i) {
        int pp = i * THREADS + tid;
        float v = s[pp];
        if (v > CONF_T) {
          unsigned long long key =
              ((unsigned long long)__float_as_uint(v) << 32) |
              (unsigned long long)(0xFFFFFFFFu - (unsigned)pp);
          if (key > nk) nk = key;
        }
      }
      curkey = nk;
    }
    // next round's wavemax writes are fenced by the first barrier above
  }
  __syncthreads();

  // ---- Gather candidate boxes + areas ----
  if (tid < TOPK) {
    int p = cand_i[tid];
    if (p >= 0) {
      const float4 bb =
          reinterpret_cast<const float4*>(boxes)[(size_t)b * PRIORS + p];
      bx[tid * 4 + 0] = bb.x;
      bx[tid * 4 + 1] = bb.y;
      bx[tid * 4 + 2] = bb.z;
      bx[tid * 4 + 3] = bb.w;
      area[tid] = fmaxf(bb.z - bb.x, 0.0f) * fmaxf(bb.w - bb.y, 0.0f);
      keep[tid] = 1;
    } else {
      bx[tid * 4 + 0] = 0.0f; bx[tid * 4 + 1] = 0.0f;
      bx[tid * 4 + 2] = 0.0f; bx[tid * 4 + 3] = 0.0f;
      area[tid] = 0.0f;
      keep[tid] = 0;
    }
  }
  __syncthreads();

  // ---- Greedy NMS: currently-kept j suppresses strictly-later t ----
  for (int j = 0; j < TOPK; ++j) {
    if (keep[j] && tid > j && tid < TOPK && keep[tid]) {
      float xx1 = fmaxf(bx[j * 4 + 0], bx[tid * 4 + 0]);
      float yy1 = fmaxf(bx[j * 4 + 1], bx[tid * 4 + 1]);
      float xx2 = fminf(bx[j * 4 + 2], bx[tid * 4 + 2]);
      float yy2 = fminf(bx[j * 4 + 3], bx[tid * 4 + 3]);
      float iw = fmaxf(xx2 - xx1, 0.0f);
      float ih = fmaxf(yy2 - yy1, 0.0f);
      float inter = iw * ih;
      float uni = area[j] + area[tid] - inter;
      float iou = inter / fmaxf(uni, 1e-12f);
      if (iou > NMS_T) keep[tid] = 0;
    }
    __syncthreads();
  }

  // ---- Stable compaction positions ----
  if (tid == 0) {
    int cnt = 0;
    for (int k2 = 0; k2 < TOPK; ++k2) pos[k2] = keep[k2] ? cnt++ : -1;
  }
  __syncthreads();

  float* obase = out + (((size_t)b * NCLS + c) * TOPK) * 5;
  if (tid < TOPK && pos[tid] >= 0) {
    float* o = obase + (size_t)pos[tid] * 5;
    o[0] = cand_s[tid];
    o[1] = bx[tid * 4 + 0];
    o[2] = bx[tid * 4 + 1];
    o[3] = bx[tid * 4 + 2];
    o[4] = bx[tid * 4 + 3];
  }
}

// ---------------------------------------------------------------------------
extern "C" void kernel_launch(void* const* d_in, const int* in_sizes, int n_in,
                              void* d_out, int out_size, void* d_ws, size_t ws_size,
                              hipStream_t stream) {
  const float* loc   = (const float*)d_in[0];  // [B, P, 4]
  const float* conf  = (const float*)d_in[1];  // [B*P, C]
  const float* prior = (const float*)d_in[2];  // [P, 4]
  float* out   = (float*)d_out;                // [B, C, K, 5]
  float* boxes = (float*)d_ws;                 // [B*P, 4] scratch (8 MB)

  zero_out_kernel<<<2048, 256, 0, stream>>>(out, out_size);

  decode_kernel<<<(BATCH * PRIORS + 255) / 256, 256, 0, stream>>>(loc, prior, boxes);

  constexpr size_t SMEM =
      (size_t)PRIORS * sizeof(float)                  // scores
      + (NWAVES + 1) * sizeof(unsigned long long)     // wavemax + winner
      + TOPK * sizeof(float)                          // cand_s
      + TOPK * sizeof(int)                            // cand_i
      + TOPK * 4 * sizeof(float)                      // bx
      + TOPK * sizeof(float)                          // area
      + TOPK * sizeof(int)                            // keep
      + TOPK * sizeof(int);                           // pos
  detect_kernel<<<BATCH * (NCLS - 1), THREADS, SMEM, stream>>>(conf, boxes, out);
}